// Quantize_30803505447135
// MI455X (gfx1250) — compile-verified
//
#include <hip/hip_runtime.h>
#include <stdint.h>

// Problem constants (match reference)
#define N_OUT 4096
#define N_IN  11008
#define DSUB  8
#define KC    256
#define SUBS_PER_ROW (N_IN / DSUB)   // 1376
#define CHUNK 1024                   // columns per block = 256 threads * 4 floats
#define ROWS_PER_BLOCK 16

typedef float v4f __attribute__((ext_vector_type(4)));

__global__ __launch_bounds__(256) void Quantize_dequant_kernel(
    const float* __restrict__ centroids,     // [KC * DSUB]  (8 KB)
    const int*   __restrict__ assignments,   // [N_OUT * SUBS_PER_ROW] (codes 0..255)
    const float* __restrict__ rowwise,       // [N_IN]  per-column scale
    const float* __restrict__ colwise,       // [N_OUT] per-row scale
    float*       __restrict__ out)           // [N_OUT * N_IN]
{
    __shared__ float s_cent[KC * DSUB];      // 8 KB centroid table
    __shared__ float s_rn[CHUNK];            // 4 KB rowwise-norm slice

    const int tid     = threadIdx.x;
    const int colBase = blockIdx.x * CHUNK;
    const int rowBase = blockIdx.y * ROWS_PER_BLOCK;

    // ---- Stage centroid table into LDS via CDNA5 async copy ----
    // 256 lanes x 16B = 4 KB per instruction; offset applies to BOTH LDS and
    // global addresses (CDNA5 ISA pseudocode), so two issues
    // with the same base regs cover all 8 KB.
    {
        uint32_t lds0 = (uint32_t)(uintptr_t)(&s_cent[0]) + (uint32_t)tid * 16u;
        uint64_t g0   = (uint64_t)(uintptr_t)(centroids)  + (uint64_t)tid * 16u;
        asm volatile("global_load_async_to_lds_b128 %0, %1, off"
                     :: "v"(lds0), "v"(g0) : "memory");
        asm volatile("global_load_async_to_lds_b128 %0, %1, off offset:4096"
                     :: "v"(lds0), "v"(g0) : "memory");
    }

    // ---- Stage this block's rowwise-norm slice (16B per lane) ----
    const int j = colBase + tid * 4;         // first output column for this lane
    const bool inb = (j < N_IN);
    if (inb) {
        uint32_t ldsr = (uint32_t)(uintptr_t)(&s_rn[0]) + (uint32_t)tid * 16u;
        uint64_t gr   = (uint64_t)(uintptr_t)(rowwise + j);
        asm volatile("global_load_async_to_lds_b128 %0, %1, off"
                     :: "v"(ldsr), "v"(gr) : "memory");
    }

    asm volatile("s_wait_asynccnt 0" ::: "memory");
    __syncthreads();

    if (!inb) return;                        // tail column chunk: idle lanes done

    const v4f rn4 = *reinterpret_cast<const v4f*>(&s_rn[tid * 4]);
    const int sub = j >> 3;                  // which 8-wide subvector this lane reads
    const int halfOff = (tid & 1) ? 4 : 0;   // which half of the codeword (floats)

    #pragma unroll
    for (int r = 0; r < ROWS_PER_BLOCK; ++r) {
        const int row = rowBase + r;
        const float cn = colwise[row];       // uniform -> s_load_b32

        const int code = assignments[(size_t)row * SUBS_PER_ROW + sub];
        const v4f c4 = *reinterpret_cast<const v4f*>(&s_cent[code * DSUB + halfOff]);

        v4f o;
        o.x = c4.x * rn4.x * cn;
        o.y = c4.y * rn4.y * cn;
        o.z = c4.z * rn4.z * cn;
        o.w = c4.w * rn4.w * cn;

        // Streaming store: 180 MB output shouldn't thrash L2 (keep the
        // 22 MB assignments stream resident across graph replays).
        __builtin_nontemporal_store(o,
            reinterpret_cast<v4f*>(out + (size_t)row * N_IN + j));
    }
}

extern "C" void kernel_launch(void* const* d_in, const int* in_sizes, int n_in,
                              void* d_out, int out_size, void* d_ws, size_t ws_size,
                              hipStream_t stream) {
    (void)in_sizes; (void)n_in; (void)out_size; (void)d_ws; (void)ws_size;

    const float* centroids   = (const float*)d_in[0];
    const int*   assignments = (const int*)  d_in[1];  // harness: integer -> const int*
    const float* rowwise     = (const float*)d_in[2];
    const float* colwise     = (const float*)d_in[3];
    float*       out         = (float*)d_out;

    dim3 grid((N_IN + CHUNK - 1) / CHUNK,   // 11 column chunks
              N_OUT / ROWS_PER_BLOCK);      // 256 row groups
    Quantize_dequant_kernel<<<grid, 256, 0, stream>>>(
        centroids, assignments, rowwise, colwise, out);
}